// yolo_19636590477732
// MI455X (gfx1250) — compile-verified
//
#include <hip/hip_runtime.h>
#include <stdint.h>

// ---------------- problem constants ----------------
#define BB 32
#define SS 52
#define NCLS 80
#define FEAT 85                      // 5 + 80, contiguous per cell after reshape
#define CELLS (BB * 3 * SS * SS)     // 259584
#define CPC 64                       // cells per TDM tile
#define CHUNK_FLOATS (CPC * FEAT)    // 5440  (fits 16-bit tile_dim0)
#define CHUNK_BYTES (CHUNK_FLOATS * 4)
#define NCHUNKS (CELLS / CPC)        // 4056 (exact)
#define NBLK 507
#define CHUNKS_PER_BLK (NCHUNKS / NBLK) // 8 (exact)
#define MAXG 256

// anchors[MASK]/stride(=8) in grid units
__device__ __constant__ float c_aw[3] = {1.25f, 2.0f, 4.125f};
__device__ __constant__ float c_ah[3] = {1.625f, 3.75f, 2.875f};

typedef unsigned int u32x4 __attribute__((ext_vector_type(4)));
typedef int i32x4 __attribute__((ext_vector_type(4)));
typedef int i32x8 __attribute__((ext_vector_type(8)));

// ---------------- CDNA5 TDM: mandatory on the device pass ----------------
#if defined(__HIP_DEVICE_COMPILE__)
#define DEV_TDM 1
#else
#define DEV_TDM 0
#endif

#if DEV_TDM
// Issue one TDM load of CHUNK_BYTES contiguous bytes: global -> LDS.
// D# group0: count=1 | lds_addr | global_addr[56:0] | type=2
// D# group1: data_size=4B, 1-D tile: tensor_dim0=tile_dim0=CHUNK_FLOATS, dims1..2 trivial
__device__ __forceinline__ void tdm_issue(void* ldsptr, uint64_t gaddr) {
  u32x4 g0;
  g0.x = 1u;                                        // count=1, no gather, not restore
  g0.y = (uint32_t)(uintptr_t)ldsptr;               // lds_addr (bytes; flat addr low 32 = LDS offset)
  g0.z = (uint32_t)gaddr;                           // global_addr[31:0]
  g0.w = ((uint32_t)(gaddr >> 32) & 0x01FFFFFFu) | 0x80000000u; // addr[56:32] | type=2
  i32x8 g1 = { 0x00020000,                          // data_size=2 (4 bytes), mask/flags=0
               (int)((unsigned)CHUNK_FLOATS << 16), // tensor_dim0[15:0] in [63:48]
               (1 << 16),                           // tensor_dim0 hi=0 | tensor_dim1 lo=1
               (int)((unsigned)CHUNK_FLOATS << 16), // tensor_dim1 hi=0 | tile_dim0
               1,                                   // tile_dim1=1, tile_dim2=0
               CHUNK_FLOATS,                        // tensor_dim0_stride lo32
               0, 0 };
#if __has_builtin(__builtin_amdgcn_tensor_load_to_lds_d2)
  __builtin_amdgcn_tensor_load_to_lds_d2(g0, g1, 0);
#elif __has_builtin(__builtin_amdgcn_tensor_load_to_lds)
#if defined(__clang_major__) && __clang_major__ >= 23
  __builtin_amdgcn_tensor_load_to_lds(g0, g1, (i32x4)0, (i32x4)0, (i32x8)0, 0);
#else
  __builtin_amdgcn_tensor_load_to_lds(g0, g1, (i32x4)0, (i32x4)0, 0);
#endif
#else
  // Portable last resort per bridge doc: raw VIMAGE 2-group form.
  asm volatile("tensor_load_to_lds %0, %1" :: "s"(g0), "s"(g1) : "memory");
#endif
}
#if __has_builtin(__builtin_amdgcn_s_wait_tensorcnt)
#define TDM_WAIT0() __builtin_amdgcn_s_wait_tensorcnt(0)
#define TDM_WAIT1() __builtin_amdgcn_s_wait_tensorcnt(1)
#else
#define TDM_WAIT0() asm volatile("s_wait_tensorcnt 0x0" ::: "memory")
#define TDM_WAIT1() asm volatile("s_wait_tensorcnt 0x1" ::: "memory")
#endif
#endif // DEV_TDM

// ---------------- math helpers ----------------
// softplus(x) = -log(1-sigmoid(x)) = -log(sigmoid(-x)); BCE log-clamp => min(.,100)
__device__ __forceinline__ float softplusf(float x) {
  if (x > 20.f) return x;
  if (x < -20.f) return __expf(x);
  return __logf(1.f + __expf(x));
}
__device__ __forceinline__ float sigm(float x) { return 1.f / (1.f + __expf(-x)); }

// acc layout in d_ws (floats):
// 0 lossx 1 lossy 2 lossw 3 lossh 4 obj 5 cls 6 iou_num 7 nmask 8 r50 9 r75 10 ncells
__global__ void k_init(float* acc) {
  if (threadIdx.x < 16) acc[threadIdx.x] = 0.f;
}

// ---------------- dense pass: stream 88MB via TDM, sum min(softplus,100) over feats 4..84
__global__ void __launch_bounds__(256) k_dense(const float* __restrict__ pred,
                                               float* __restrict__ acc) {
  __shared__ float lbuf[2][CHUNK_FLOATS];   // 2 x 21,760 B double buffer
  __shared__ float red[2][8];
  const int tid = threadIdx.x;
  const int wid = tid >> 5;
  const int lane = tid & 31;
  const int base = blockIdx.x * CHUNKS_PER_BLK;
  float accObj = 0.f, accCls = 0.f;

#if DEV_TDM
  if (wid == 0) {
    tdm_issue(&lbuf[0][0],
              (uint64_t)(uintptr_t)pred + (uint64_t)base * CHUNK_BYTES);
  }
#endif
  for (int i = 0; i < CHUNKS_PER_BLK; ++i) {
    const int par = i & 1;
#if DEV_TDM
    if (wid == 0) {
      if (i + 1 < CHUNKS_PER_BLK) {
        tdm_issue(&lbuf[(i + 1) & 1][0],
                  (uint64_t)(uintptr_t)pred + (uint64_t)(base + i + 1) * CHUNK_BYTES);
        TDM_WAIT1();   // chunk i complete (TDM ops are in-order per wave)
      } else {
        TDM_WAIT0();
      }
    }
#else
    // Host-pass placeholder only (semantic checking); device always uses TDM.
    for (int idx = tid; idx < CHUNK_FLOATS; idx += 256)
      lbuf[par][idx] = pred[(size_t)(base + i) * CHUNK_FLOATS + idx];
#endif
    __syncthreads();
    const volatile float* vb = &lbuf[par][0];
    // 8 waves x 8 cells = 64 cells; lanes cover feats 4..84 stride-1 (bank-conflict free)
    for (int j = 0; j < 8; ++j) {
      const int off = (wid * 8 + j) * FEAT;
      float t0 = fminf(softplusf(vb[off + 4 + lane]), 100.f);     // feats 4..35
      if (lane == 0) accObj += t0; else accCls += t0;
      accCls += fminf(softplusf(vb[off + 36 + lane]), 100.f);     // feats 36..67
      if (lane < 17)
        accCls += fminf(softplusf(vb[off + 68 + lane]), 100.f);   // feats 68..84
    }
    __syncthreads();
  }
  // wave32 shuffle reduce, then one atomicAdd pair per block
  for (int o = 16; o; o >>= 1) {
    accObj += __shfl_down(accObj, o, 32);
    accCls += __shfl_down(accCls, o, 32);
  }
  if (lane == 0) { red[0][wid] = accObj; red[1][wid] = accCls; }
  __syncthreads();
  if (tid == 0) {
    float a = 0.f, c = 0.f;
    for (int w = 0; w < 8; ++w) { a += red[0][w]; c += red[1][w]; }
    atomicAdd(&acc[4], a);
    atomicAdd(&acc[5], c);
  }
}

// ---------------- sparse pass: 200 GT x 3 anchors ----------------
__global__ void k_sparse(const float* __restrict__ pred, const float* __restrict__ gt,
                         int G, float* __restrict__ acc) {
  __shared__ float scx[MAXG], scy[MAXG], sgw[MAXG], sgh[MAXG], ssc[MAXG];
  __shared__ int six[MAXG], siy[MAXG], sbb[MAXG], slab[MAXG];
  __shared__ int sm[3 * MAXG];
  const int t = threadIdx.x;
  if (t < G) {
    slab[t] = (int)gt[t * 6 + 0];
    float cx = gt[t * 6 + 1] * 52.f, cy = gt[t * 6 + 2] * 52.f;
    float gw = gt[t * 6 + 3] * 52.f, gh = gt[t * 6 + 4] * 52.f;
    sbb[t] = (int)gt[t * 6 + 5];
    scx[t] = cx; scy[t] = cy; sgw[t] = gw; sgh[t] = gh;
    six[t] = (int)floorf(cx); siy[t] = (int)floorf(cy);
    ssc[t] = 2.f - (gw * 8.f) * (gh * 8.f) / (416.f * 416.f);
  }
  __syncthreads();
  if (t < 3 * G) {
    int g = t / 3, a = t % 3;
    float cx = scx[g], cy = scy[g], gw = sgw[g], gh = sgh[g];
    float aw = c_aw[a], ah = c_ah[a];
    float iw = fmaxf(0.f, fminf(cx + gw * .5f, cx + aw * .5f) - fmaxf(cx - gw * .5f, cx - aw * .5f));
    float ih = fmaxf(0.f, fminf(cy + gh * .5f, cy + ah * .5f) - fmaxf(cy - gh * .5f, cy - ah * .5f));
    float inter = iw * ih;
    float iou = inter / (gw * gh + aw * ah - inter);
    sm[t] = (iou > 0.3f) ? 1 : 0;
    // warm the gather rows early (gfx1250 global_prefetch_b8)
    size_t basei = ((size_t)((sbb[g] * 3 + a) * 2704 + six[g] * 52 + siy[g])) * FEAT;
    __builtin_prefetch(pred + basei, 0, 1);
  }
  __syncthreads();
  if (t < 3 * G) {
    int g = t / 3, a = t % 3;
    int b = sbb[g], ix = six[g], iy = siy[g];
    size_t basei = ((size_t)((b * 3 + a) * 2704 + ix * 52 + iy)) * FEAT;
    float p0 = pred[basei + 0], p1 = pred[basei + 1];
    float p2 = pred[basei + 2], p3 = pred[basei + 3], p4 = pred[basei + 4];
    float aw = c_aw[a], ah = c_ah[a];
    float cx = scx[g], cy = scy[g], gw = sgw[g], gh = sgh[g], sc = ssc[g];
    // decoded prediction at the assigned cell (x along H axis per reference)
    float px = sigm(p0) + (float)iy;
    float py = sigm(p1) + (float)ix;
    float pw = __expf(p2) * aw, ph = __expf(p3) * ah;
    float iw = fmaxf(0.f, fminf(cx + gw * .5f, px + pw * .5f) - fmaxf(cx - gw * .5f, px - pw * .5f));
    float ih = fmaxf(0.f, fminf(cy + gh * .5f, py + ph * .5f) - fmaxf(cy - gh * .5f, py - ph * .5f));
    float inter = iw * ih;
    float piou = inter / (gw * gh + pw * ph - inter);
    if (sm[t]) {
      atomicAdd(&acc[6], (1.f - piou) * sc);
      atomicAdd(&acc[7], 1.f);
      if (piou > 0.5f)  atomicAdd(&acc[8], 1.f);
      if (piou > 0.75f) atomicAdd(&acc[9], 1.f);
      // scatter dedup: last (highest g) matched writer to this (b,a,ix,iy) wins
      bool win = true;
      for (int g2 = g + 1; g2 < G; ++g2)
        if (sm[g2 * 3 + a] && six[g2] == ix && siy[g2] == iy && sbb[g2] == b) { win = false; break; }
      if (win) {
        float tx = cx - floorf(cx), ty = cy - floorf(cy);
        float bx = tx * fminf(softplusf(-p0), 100.f) + (1.f - tx) * fminf(softplusf(p0), 100.f);
        float by = ty * fminf(softplusf(-p1), 100.f) + (1.f - ty) * fminf(softplusf(p1), 100.f);
        atomicAdd(&acc[0], sc * bx);
        atomicAdd(&acc[1], sc * by);
        float ow = __logf(gw / aw), oh = __logf(gh / ah);
        float dw = p2 * sc - ow * sc, dh = p3 * sc - oh * sc;
        atomicAdd(&acc[2], dw * dw);
        atomicAdd(&acc[3], dh * dh);
        // obj + label-class corrections: baseline (t=0) was added in dense pass
        atomicAdd(&acc[4], fminf(softplusf(-p4), 100.f) - fminf(softplusf(p4), 100.f));
        float pc = pred[basei + 5 + slab[g]];
        atomicAdd(&acc[5], fminf(softplusf(-pc), 100.f) - fminf(softplusf(pc), 100.f));
        atomicAdd(&acc[10], 1.f);   // classes.sum() denominator
      }
    }
  }
}

__global__ void k_final(const float* __restrict__ acc, float* __restrict__ out) {
  if (threadIdx.x == 0 && blockIdx.x == 0) {
    const float N1 = (float)CELLS;
    float lx = acc[0] / N1, ly = acc[1] / N1, lw = acc[2] / N1, lh = acc[3] / N1;
    float lo = acc[4] / N1, lc = acc[5] / (N1 * (float)NCLS);
    float nm = acc[7];
    float li = (nm > 0.f) ? acc[6] / fmaxf(nm, 1.f) : 0.f;
    float denom = acc[10] + 1e-10f;
    float r50 = acc[8] / denom, r75 = acc[9] / denom;
    float loss = lx + ly + lw + lh + lo + lc + li;
    out[0] = loss; out[1] = lx; out[2] = ly; out[3] = lw; out[4] = lh;
    out[5] = lo;  out[6] = lc; out[7] = li; out[8] = r50; out[9] = r75;
  }
}

extern "C" void kernel_launch(void* const* d_in, const int* in_sizes, int n_in,
                              void* d_out, int out_size, void* d_ws, size_t ws_size,
                              hipStream_t stream) {
  (void)n_in; (void)out_size; (void)ws_size;
  const float* pred = (const float*)d_in[0];
  const float* gt   = (const float*)d_in[1];
  int G = in_sizes[1] / 6;
  if (G > MAXG) G = MAXG;
  float* acc = (float*)d_ws;
  float* out = (float*)d_out;

  k_init<<<1, 32, 0, stream>>>(acc);
  k_dense<<<NBLK, 256, 0, stream>>>(pred, acc);
  k_sparse<<<1, 3 * MAXG, 0, stream>>>(pred, gt, G, acc);
  k_final<<<1, 1, 0, stream>>>(acc, out);
}